// Linear_act_sp_15264313770241
// MI455X (gfx1250) — compile-verified
//
#include <hip/hip_runtime.h>
#include <stdint.h>

// ---------------------------------------------------------------------------
// Linear with 2:4 activation pruning + variance correction, gfx1250.
//   out = v * prune24(x) @ W^T,  v = sqrt(var(x)/max(var(prune24(x)),1e-9))
// GEMM uses v_swmmac_f32_16x16x64_bf16 (2:4 structured sparse, half-size A +
// 2-bit indices, ISA 7.12.3/7.12.4), tiles staged with async global->LDS
// copies (ASYNCcnt).
// ---------------------------------------------------------------------------

typedef __attribute__((ext_vector_type(16))) __bf16 v16bf;
typedef __attribute__((ext_vector_type(32))) __bf16 v32bf;
typedef __attribute__((ext_vector_type(8)))  float  v8f;

static constexpr int Mdim = 16384;
static constexpr int Ndim = 2048;
static constexpr int Kdim = 2048;

union ABFrag {   // SWMMAC A fragment: 16x32 packed bf16 (expands to 16x64)
  v16bf v;
  uint4 u[2];
};
union BFrag {    // SWMMAC B fragment: 64x16 bf16 = 32 bf16/lane (16 VGPRs)
  v32bf v;
  uint4 u[4];
};

// round-to-nearest-even f32 -> bf16 bits
__device__ __forceinline__ unsigned int f2bf(float f) {
  unsigned int u = __float_as_uint(f);
  u += 0x7FFFu + ((u >> 16) & 1u);
  return (u >> 16) & 0xFFFFu;
}

// ----- async global->LDS copies (CDNA5, ASYNCcnt) with safe fallback -------
#if __has_builtin(__builtin_amdgcn_global_load_async_to_lds_b128) && \
    __has_builtin(__builtin_amdgcn_global_load_async_to_lds_b32)
#define HAVE_ASYNC_LDS 1
#else
#define HAVE_ASYNC_LDS 0
#endif

#if HAVE_ASYNC_LDS
typedef int v4i __attribute__((vector_size(16)));
typedef __attribute__((address_space(1))) v4i* gptr_v4i;
typedef __attribute__((address_space(3))) v4i* lptr_v4i;
typedef __attribute__((address_space(1))) int* gptr_i32;
typedef __attribute__((address_space(3))) int* lptr_i32;

__device__ __forceinline__ void async_cp_b128(const void* g, void* l) {
  gptr_v4i gp = (gptr_v4i)(uintptr_t)g;
  lptr_v4i lp = (lptr_v4i)(uint32_t)(uintptr_t)l;
  __builtin_amdgcn_global_load_async_to_lds_b128(gp, lp, 0, 0);
}
__device__ __forceinline__ void async_cp_b32(const void* g, void* l) {
  gptr_i32 gp = (gptr_i32)(uintptr_t)g;
  lptr_i32 lp = (lptr_i32)(uint32_t)(uintptr_t)l;
  __builtin_amdgcn_global_load_async_to_lds_b32(gp, lp, 0, 0);
}
__device__ __forceinline__ void wait_async0() {
#if __has_builtin(__builtin_amdgcn_s_wait_asynccnt)
  __builtin_amdgcn_s_wait_asynccnt(0);
#else
  asm volatile("s_wait_asynccnt 0x0" ::: "memory");
#endif
}
#endif

// ---------------------------------------------------------------------------
// Kernel 1: 2:4 prune -> packed SWMMAC operands + variance statistics.
// One thread = 8 groups of 4 (32 elems) so each index dword has one writer.
//   apk : [M][K/2] bf16, two kept values per group in ascending-index order
//   aidx: [M][K/32] u32, group g in bits [4g+3:4g] (idx0 low, idx1 high)
// ---------------------------------------------------------------------------
__global__ __launch_bounds__(256) void prune_pack_stats_kernel(
    const float4* __restrict__ x4,
    uint4* __restrict__ apk,
    unsigned int* __restrict__ aidx,
    double* __restrict__ stats,   // [sum_x, sumsq_x, sum_sp, sumsq_sp]
    int nthreads)
{
  const int tid = threadIdx.x;
  const int t = blockIdx.x * blockDim.x + tid;

  double p0 = 0.0, p1 = 0.0, p2 = 0.0, p3 = 0.0;
  if (t < nthreads) {
    float sx = 0.f, ssx = 0.f, sp = 0.f, ssp = 0.f;
    unsigned int idxword = 0;
    unsigned int pw[8];
#pragma unroll
    for (int g = 0; g < 8; ++g) {
      float4 q = x4[(size_t)t * 8 + g];
      float a[4] = {q.x, q.y, q.z, q.w};
      float m[4];
#pragma unroll
      for (int i = 0; i < 4; ++i) m[i] = fabsf(a[i]);

      unsigned int v0 = 0, v1 = 0, code = 0;
      int slot = 0;
#pragma unroll
      for (int i = 0; i < 4; ++i) {
        int rank = 0;
#pragma unroll
        for (int j = 0; j < 4; ++j)
          rank += (m[j] > m[i]) || ((m[j] == m[i]) && (j < i));
        const bool keep = (rank < 2);   // exactly 2 kept (ranks form a permutation)
        const float kept = keep ? a[i] : 0.0f;
        sx += a[i]; ssx += a[i] * a[i];
        sp += kept; ssp += kept * kept;
        if (keep) {
          code |= (unsigned int)i << (2 * slot);   // ascending => Idx0 < Idx1
          if (slot == 0) v0 = f2bf(a[i]); else v1 = f2bf(a[i]);
          ++slot;
        }
      }
      pw[g] = v0 | (v1 << 16);
      idxword |= code << (4 * g);
    }
    uint4 o0; o0.x = pw[0]; o0.y = pw[1]; o0.z = pw[2]; o0.w = pw[3];
    uint4 o1; o1.x = pw[4]; o1.y = pw[5]; o1.z = pw[6]; o1.w = pw[7];
    apk[(size_t)t * 2 + 0] = o0;
    apk[(size_t)t * 2 + 1] = o1;
    aidx[t] = idxword;
    p0 = sx; p1 = ssx; p2 = sp; p3 = ssp;
  }

#pragma unroll
  for (int off = 16; off > 0; off >>= 1) {
    p0 += __shfl_down(p0, off, 32);
    p1 += __shfl_down(p1, off, 32);
    p2 += __shfl_down(p2, off, 32);
    p3 += __shfl_down(p3, off, 32);
  }
  __shared__ double red[8][4];
  const int wave = tid >> 5, lane = tid & 31;
  if (lane == 0) {
    red[wave][0] = p0; red[wave][1] = p1; red[wave][2] = p2; red[wave][3] = p3;
  }
  __syncthreads();
  if (tid == 0) {
    double t0 = 0, t1 = 0, t2 = 0, t3 = 0;
    for (int w = 0; w < 8; ++w) {
      t0 += red[w][0]; t1 += red[w][1]; t2 += red[w][2]; t3 += red[w][3];
    }
    atomicAdd(&stats[0], t0);
    atomicAdd(&stats[1], t1);
    atomicAdd(&stats[2], t2);
    atomicAdd(&stats[3], t3);
  }
}

// ---------------------------------------------------------------------------
// Kernel 2: v = sqrt(var(x,ddof=1) / max(var(x_sp,ddof=1), 1e-9))
// ---------------------------------------------------------------------------
__global__ void finalize_v_kernel(const double* __restrict__ stats,
                                  float* __restrict__ vout)
{
  const double n = (double)Mdim * (double)Kdim;
  double sx = stats[0], ssx = stats[1], sp = stats[2], ssp = stats[3];
  double varx = (ssx - sx * sx / n) / (n - 1.0);
  double varp = (ssp - sp * sp / n) / (n - 1.0);
  varp = fmax(varp, 1e-9);
  *vout = (float)sqrt(varx / varp);
}

// ---------------------------------------------------------------------------
// Kernel 3: weight f32 -> bf16
// ---------------------------------------------------------------------------
__global__ __launch_bounds__(256) void convert_w_kernel(
    const float4* __restrict__ w4,
    unsigned long long* __restrict__ out,
    int n4)
{
  int i = blockIdx.x * blockDim.x + threadIdx.x;
  if (i < n4) {
    float4 q = w4[i];
    unsigned long long pk =
        (unsigned long long)f2bf(q.x)         |
        ((unsigned long long)f2bf(q.y) << 16) |
        ((unsigned long long)f2bf(q.z) << 32) |
        ((unsigned long long)f2bf(q.w) << 48);
    out[i] = pk;
  }
}

// ---------------------------------------------------------------------------
// Kernel 4: C[M,N] = v * expand24(Ap,Ai)[M,K] x B[N,K]^T via SWMMAC bf16.
// 128x128 block tile, 8 waves of 64x32 (4x2 frags), K-step = 64 expanded.
// ---------------------------------------------------------------------------
#define BM 128
#define BN 128
#define BKE 64        // expanded K per step (packed: 32)
#define ASTR 40       // packed-A LDS row stride (ushorts): 80B
#define BSTR 72       // B LDS row stride (ushorts): 144B

__global__ __launch_bounds__(256) void gemm_swmmac_bf16_kernel(
    const unsigned short* __restrict__ Ap,  // packed A [Mdim][Kdim/2] bf16
    const unsigned int*  __restrict__ Ai,   // indices  [Mdim][Kdim/32] u32
    const unsigned short* __restrict__ B,   // weight bf16 [Ndim][Kdim]
    const float* __restrict__ vscale,
    float* __restrict__ C)                  // [Mdim][Ndim] f32
{
  __shared__ __align__(16) unsigned short As[2][BM][ASTR];
  __shared__ __align__(16) unsigned short Bs[2][BN][BSTR];
  __shared__ unsigned int Is[2][BM][2];

  const int tid  = threadIdx.x;
  const int wave = tid >> 5;
  const int lane = tid & 31;
  const int hi   = lane >> 4;
  const int lr   = lane & 15;

  const int m0 = blockIdx.y * BM;
  const int n0 = blockIdx.x * BN;
  const int wm = (wave >> 2) * 64;
  const int wn = (wave & 3) * 32;

  const size_t Kp = Kdim / 2;    // 1024 packed cols
  const size_t Ki = Kdim / 32;   // 64 index dwords per row
  const int KT = Kdim / BKE;     // 32 K-steps

  v8f acc[4][2];
#pragma unroll
  for (int i = 0; i < 4; ++i)
#pragma unroll
    for (int j = 0; j < 2; ++j)
#pragma unroll
      for (int r = 0; r < 8; ++r) acc[i][j][r] = 0.0f;

  // staging decomposition
  const int ca    = tid * 2;                 // A: 512 x 16B chunks, 2/thread
  const int arow  = ca >> 2;
  const int acol0 = (ca & 3) * 8;
  const int acol1 = ((ca + 1) & 3) * 8;
  const int brow  = tid >> 1;                // B: 1024 x 16B chunks, 4/thread
  const int bcol  = (tid & 1) * 32;
  const int irow  = tid >> 1;                // I: 256 dwords, 1/thread
  const int iw    = tid & 1;

  const unsigned short* gA = &Ap[(size_t)(m0 + arow) * Kp];
  const unsigned short* gB = &B[(size_t)(n0 + brow) * Kdim + bcol];
  const unsigned int*   gI = &Ai[(size_t)(m0 + irow) * Ki + iw];

#if HAVE_ASYNC_LDS
  auto stage = [&](int buf, int kb) {
    const unsigned short* sa = gA + (size_t)kb * 32;
    async_cp_b128(sa + acol0, &As[buf][arow][acol0]);
    async_cp_b128(sa + acol1, &As[buf][arow][acol1]);
    const unsigned short* sb = gB + (size_t)kb * 64;
    async_cp_b128(sb,      &Bs[buf][brow][bcol]);
    async_cp_b128(sb + 8,  &Bs[buf][brow][bcol + 8]);
    async_cp_b128(sb + 16, &Bs[buf][brow][bcol + 16]);
    async_cp_b128(sb + 24, &Bs[buf][brow][bcol + 24]);
    async_cp_b32(gI + (size_t)kb * 2, &Is[buf][irow][iw]);
  };
  stage(0, 0);
  wait_async0();
  __syncthreads();
#else
  uint4 ra[2], rb[4];
  unsigned int ri;
  auto g2r = [&](int kb) {
    const unsigned short* sa = gA + (size_t)kb * 32;
    ra[0] = *(const uint4*)(sa + acol0);
    ra[1] = *(const uint4*)(sa + acol1);
    const unsigned short* sb = gB + (size_t)kb * 64;
    rb[0] = *(const uint4*)(sb);
    rb[1] = *(const uint4*)(sb + 8);
    rb[2] = *(const uint4*)(sb + 16);
    rb[3] = *(const uint4*)(sb + 24);
    ri = gI[(size_t)kb * 2];
  };
  auto r2l = [&](int buf) {
    *(uint4*)&As[buf][arow][acol0] = ra[0];
    *(uint4*)&As[buf][arow][acol1] = ra[1];
    *(uint4*)&Bs[buf][brow][bcol]      = rb[0];
    *(uint4*)&Bs[buf][brow][bcol + 8]  = rb[1];
    *(uint4*)&Bs[buf][brow][bcol + 16] = rb[2];
    *(uint4*)&Bs[buf][brow][bcol + 24] = rb[3];
    Is[buf][irow][iw] = ri;
  };
  g2r(0);
  r2l(0);
  __syncthreads();
#endif

  // cap unrolling at 2: keeps buf compile-time per half, one persistent acc
  // register set (avoids full-unroll code bloat + acc copy movs)
#pragma clang loop unroll_count(2)
  for (int kb = 0; kb < KT; ++kb) {
    const int buf = kb & 1;
    const bool more = (kb + 1) < KT;

#if HAVE_ASYNC_LDS
    if (more) stage(buf ^ 1, kb + 1);   // buf^1 readers finished at last barrier
#else
    if (more) g2r(kb + 1);
#endif
    if (kb + 2 < KT) {  // warm near caches for the tile after next
      __builtin_prefetch(gA + (size_t)(kb + 2) * 32, 0, 3);
      __builtin_prefetch(gB + (size_t)(kb + 2) * 64, 0, 3);
    }

    // A fragments (packed 16x32): lanes 0-15: p0..7 & p16..23, lanes 16-31: p8..15 & p24..31
    ABFrag af[4];
    unsigned int ix[4];
#pragma unroll
    for (int i = 0; i < 4; ++i) {
      const int r = wm + i * 16 + lr;
      af[i].u[0] = *(const uint4*)&As[buf][r][hi ? 8 : 0];
      af[i].u[1] = *(const uint4*)&As[buf][r][hi ? 24 : 16];
      ix[i] = Is[buf][r][hi];
    }
    // j-outer: only one 16-VGPR B fragment live at a time
#pragma unroll
    for (int j = 0; j < 2; ++j) {
      const int r = wn + j * 16 + lr;
      const unsigned short* bp = &Bs[buf][r][0];
      const int b0 = hi ? 16 : 0;
      BFrag bfr;
      bfr.u[0] = *(const uint4*)(bp + b0);
      bfr.u[1] = *(const uint4*)(bp + b0 + 8);
      bfr.u[2] = *(const uint4*)(bp + b0 + 32);
      bfr.u[3] = *(const uint4*)(bp + b0 + 40);
#pragma unroll
      for (int i = 0; i < 4; ++i)
        acc[i][j] = __builtin_amdgcn_swmmac_f32_16x16x64_bf16(
            false, af[i].v, false, bfr.v, acc[i][j], (int)ix[i], false, false);
    }

#if HAVE_ASYNC_LDS
    if (more) wait_async0();
#else
    if (more) r2l(buf ^ 1);
#endif
    __syncthreads();
  }

  // epilogue: C layout per ISA: VGPR r -> M = r + 8*hi, N = lane%16
  const float vs = *vscale;
#pragma unroll
  for (int i = 0; i < 4; ++i) {
    const int mb = m0 + wm + i * 16 + hi * 8;
#pragma unroll
    for (int j = 0; j < 2; ++j) {
      const int col = n0 + wn + j * 16 + lr;
#pragma unroll
      for (int r = 0; r < 8; ++r)
        C[(size_t)(mb + r) * Ndim + col] = vs * acc[i][j][r];
    }
  }
}

// ---------------------------------------------------------------------------
// Launch
// ---------------------------------------------------------------------------
extern "C" void kernel_launch(void* const* d_in, const int* in_sizes, int n_in,
                              void* d_out, int out_size, void* d_ws, size_t ws_size,
                              hipStream_t stream) {
  const float* x = (const float*)d_in[0];       // [4,4096,2048] f32
  const float* w = (const float*)d_in[1];       // [2048,2048] f32
  float* out = (float*)d_out;                   // [4,4096,2048] f32

  // workspace: [0,32) stats f64x4 | [32,36) v | [256,+8MB) W bf16
  //            [+8MB,+40MB) packed A | [+40MB,+44MB) indices
  char* ws = (char*)d_ws;
  double* stats = (double*)ws;
  float* vptr = (float*)(ws + 32);
  unsigned short* wbf = (unsigned short*)(ws + 256);
  unsigned short* apk = (unsigned short*)(ws + 256 + (size_t)Ndim * Kdim * 2);
  unsigned int*   aix = (unsigned int*)(ws + 256 + (size_t)Ndim * Kdim * 2
                                               + (size_t)Mdim * (Kdim / 2) * 2);

  (void)hipMemsetAsync(d_ws, 0, 64, stream);

  const int nthreads = Mdim * Kdim / 32;        // 1,048,576 (8 groups each)
  prune_pack_stats_kernel<<<nthreads / 256, 256, 0, stream>>>(
      (const float4*)x, (uint4*)apk, aix, stats, nthreads);

  const int n4w = Ndim * Kdim / 4;              // 1,048,576
  convert_w_kernel<<<n4w / 256, 256, 0, stream>>>(
      (const float4*)w, (unsigned long long*)wbf, n4w);

  finalize_v_kernel<<<1, 1, 0, stream>>>(stats, vptr);

  dim3 grid(Ndim / BN, Mdim / BM);              // (16, 128)
  gemm_swmmac_bf16_kernel<<<grid, 256, 0, stream>>>(apk, aix, wbf, vptr, out);
}